// HGNNModel_76038101008441
// MI455X (gfx1250) — compile-verified
//
#include <hip/hip_runtime.h>
#include <hip/hip_bf16.h>
#include <stdint.h>

typedef __attribute__((ext_vector_type(2))) float v2f;
typedef __attribute__((ext_vector_type(8))) float v8f;

#define NTOT   10240
#define NU     4096
#define EMBD   64
#define KSPLIT 8
#define KCHUNK (NTOT / KSPLIT)   // 1280
#define LDSP   76                // LDS row stride (floats): 304B, 16B-aligned,
                                 // conflict-free for both row and column reads

// ---------------------------------------------------------------------------
// Zero a float buffer (t and h accumulators live in d_ws and are poisoned).
// ---------------------------------------------------------------------------
__global__ void zero_f32(float* __restrict__ p, int n) {
    int i = blockIdx.x * blockDim.x + threadIdx.x;
    if (i < n) p[i] = 0.0f;
}

// ---------------------------------------------------------------------------
// e = par0*par1 * (diag[:,None] * ego @ W) + ego   (users rows <NU, items >=NU)
// block = 256 threads = 4 rows x 64 cols, grid = NTOT/4
// ---------------------------------------------------------------------------
__global__ void precompute_e(const float* __restrict__ ego,
                             const float* __restrict__ Wu,
                             const float* __restrict__ du,
                             const float* __restrict__ pu,
                             const float* __restrict__ Wi,
                             const float* __restrict__ di,
                             const float* __restrict__ pi,
                             float* __restrict__ e) {
    __shared__ float sh[4][EMBD];
    const int g   = threadIdx.x >> 6;
    const int c   = threadIdx.x & 63;
    const int row = blockIdx.x * 4 + g;

    const float x0 = ego[(size_t)row * EMBD + c];
    sh[g][c] = x0;
    __syncthreads();

    const float* W;
    float d, s;
    if (row < NU) { W = Wu; d = du[row];      s = pu[0] * pu[1]; }
    else          { W = Wi; d = di[row - NU]; s = pi[0] * pi[1]; }

    float acc = 0.0f;
#pragma unroll
    for (int k = 0; k < EMBD; ++k)
        acc = fmaf(sh[g][k], W[k * EMBD + c], acc);

    e[(size_t)row * EMBD + c] = s * d * acc + x0;
}

// ---------------------------------------------------------------------------
// C += op(A) @ B using V_WMMA_F32_16X16X4_F32 with double-buffered
// GLOBAL_LOAD_ASYNC_TO_LDS_B128 tile streaming (ASYNCcnt-tracked).
//   TRANSA=1 : C[m,n] += sum_k A[k,m] * B[k,n]    (t = adj^T @ e)
//   TRANSA=0 : C[m,n] += sum_k A[m,k] * B[k,n]    (h = adj  @ t)
// block = 128 threads (4 waves); each wave: 16 rows x 64 cols (4 acc tiles).
// grid = (NTOT/64, 1, KSPLIT); f32 atomic accumulate into pre-zeroed C.
// ---------------------------------------------------------------------------
template <bool TRANSA>
__global__ __launch_bounds__(128)
void gemm64(const float* __restrict__ A,
            const float* __restrict__ B,
            float* __restrict__ C) {
    __shared__ __align__(16) float tA[2][64][LDSP];
    __shared__ __align__(16) float tB[2][64][LDSP];

    const int m0    = blockIdx.x * 64;
    const int kBase = blockIdx.z * KCHUNK;
    const int kEnd  = kBase + KCHUNK;

    const int wave = threadIdx.x >> 5;   // 0..3
    const int lane = threadIdx.x & 31;
    const int lh   = lane >> 4;          // half-wave: 0 or 1
    const int ll   = lane & 15;
    const int wm   = wave * 16;          // wave's row offset inside 64-row tile

    v8f acc[4] = {};

    // Issue one 64x64 A tile + 64x64 B tile (16 b128 asyncs per thread-slot
    // pass; each wave contributes 16 async instructions -> ASYNCcnt = 16).
    auto issue_tile = [&](int buf, int k0) {
#pragma unroll
        for (int j = 0; j < 8; ++j) {
            const int i  = (int)threadIdx.x + j * 128;
            const int r  = i >> 4;          // 0..63
            const int c4 = (i & 15) << 2;   // 0,4,...,60
            const float* ga = TRANSA
                ? A + ((size_t)(k0 + r) * NTOT + (m0 + c4))
                : A + ((size_t)(m0 + r) * NTOT + (k0 + c4));
            const uint32_t la = (uint32_t)(size_t)&tA[buf][r][c4];
            asm volatile("global_load_async_to_lds_b128 %0, %1, off"
                         :: "v"(la), "v"((uint64_t)(size_t)ga) : "memory");
            const float* gb = B + ((size_t)(k0 + r) * EMBD + c4);
            const uint32_t lb = (uint32_t)(size_t)&tB[buf][r][c4];
            asm volatile("global_load_async_to_lds_b128 %0, %1, off"
                         :: "v"(lb), "v"((uint64_t)(size_t)gb) : "memory");
        }
    };

    auto compute_tile = [&](int buf) {
#pragma unroll
        for (int kk = 0; kk < 64; kk += 4) {
            // A fragment 16x4: lanes 0-15 -> K=kk+0/1, lanes 16-31 -> K=kk+2/3
            v2f a;
            if (TRANSA) {
                a.x = tA[buf][kk + 2 * lh + 0][wm + ll];
                a.y = tA[buf][kk + 2 * lh + 1][wm + ll];
            } else {
                a.x = tA[buf][wm + ll][kk + 2 * lh + 0];
                a.y = tA[buf][wm + ll][kk + 2 * lh + 1];
            }
#pragma unroll
            for (int nt = 0; nt < 4; ++nt) {
                // B 4x16: VGPR j -> K=kk+j (lanes 0-15), K=kk+2+j (lanes 16-31)
                v2f b;
                b.x = tB[buf][kk + 2 * lh + 0][nt * 16 + ll];
                b.y = tB[buf][kk + 2 * lh + 1][nt * 16 + ll];
                acc[nt] = __builtin_amdgcn_wmma_f32_16x16x4_f32(
                    false, a, false, b, (short)0, acc[nt], false, false);
            }
        }
    };

    // --- double-buffered pipeline over the K chunk -------------------------
    issue_tile(0, kBase);
    int buf = 0;
    for (int k0 = kBase; k0 + 64 < kEnd; k0 += 64) {
        issue_tile(buf ^ 1, k0 + 64);           // prefetch next tile (async)
        if (k0 + 128 < kEnd) {                  // warm L2 two tiles ahead
            const int pr = (int)threadIdx.x >> 1;
            const int pc = ((int)threadIdx.x & 1) * 32;
            const size_t poff = TRANSA
                ? ((size_t)(k0 + 128 + pr) * NTOT + (m0 + pc))
                : ((size_t)(m0 + pr) * NTOT + (k0 + 128 + pc));
            __builtin_prefetch(A + poff, 0, 1);
        }
        // Wait for the OLDER 16 asyncs (current tile) only; the 16 just
        // issued for the next tile stay in flight.
        asm volatile("s_wait_asynccnt 16" ::: "memory");
        __syncthreads();
        compute_tile(buf);
        __syncthreads();                        // all reads of buf done
        buf ^= 1;
    }
    asm volatile("s_wait_asynccnt 0" ::: "memory");
    __syncthreads();
    compute_tile(buf);

    // C/D layout: VGPR r -> M = r (lanes 0-15) / r+8 (lanes 16-31), N = ll.
#pragma unroll
    for (int nt = 0; nt < 4; ++nt) {
#pragma unroll
        for (int r = 0; r < 8; ++r) {
            const int row = m0 + wm + r + 8 * lh;
            const int col = nt * 16 + ll;
            unsafeAtomicAdd(&C[(size_t)row * EMBD + col], acc[nt][r]);
        }
    }
}

// ---------------------------------------------------------------------------
// out = (h - mean) * rsqrt(var + eps) * gamma + beta + ego
// block = 256 threads = 4 rows x 64 cols, grid = NTOT/4
// ---------------------------------------------------------------------------
__global__ void ln_residual(const float* __restrict__ h,
                            const float* __restrict__ ego,
                            const float* __restrict__ gamma,
                            const float* __restrict__ beta,
                            float* __restrict__ out) {
    __shared__ float s1[4][EMBD];
    __shared__ float s2[4][EMBD];
    const int g   = threadIdx.x >> 6;
    const int c   = threadIdx.x & 63;
    const int row = blockIdx.x * 4 + g;

    const float x = h[(size_t)row * EMBD + c];
    s1[g][c] = x;
    s2[g][c] = x * x;
    __syncthreads();
#pragma unroll
    for (int off = 32; off >= 1; off >>= 1) {
        if (c < off) {
            s1[g][c] += s1[g][c + off];
            s2[g][c] += s2[g][c + off];
        }
        __syncthreads();
    }
    const float mu  = s1[g][0] * (1.0f / EMBD);
    const float var = s2[g][0] * (1.0f / EMBD) - mu * mu;
    const float rs  = rsqrtf(var + 1e-5f);
    out[(size_t)row * EMBD + c] =
        (x - mu) * rs * gamma[c] + beta[c] + ego[(size_t)row * EMBD + c];
}

// ---------------------------------------------------------------------------
extern "C" void kernel_launch(void* const* d_in, const int* in_sizes, int n_in,
                              void* d_out, int out_size, void* d_ws, size_t ws_size,
                              hipStream_t stream) {
    const float* ego = (const float*)d_in[0];
    const float* adj = (const float*)d_in[1];
    const float* Wu  = (const float*)d_in[2];
    const float* du  = (const float*)d_in[3];
    const float* pu  = (const float*)d_in[4];
    const float* Wi  = (const float*)d_in[5];
    const float* di  = (const float*)d_in[6];
    const float* pi  = (const float*)d_in[7];
    const float* lg  = (const float*)d_in[8];
    const float* lb  = (const float*)d_in[9];
    float* out = (float*)d_out;

    float* e = (float*)d_ws;                 // NTOT x 64
    float* t = e + (size_t)NTOT * EMBD;      // NTOT x 64 (atomic accum)
    float* h = t + (size_t)NTOT * EMBD;      // NTOT x 64 (atomic accum)

    const int nZero = 2 * NTOT * EMBD;
    zero_f32<<<(nZero + 255) / 256, 256, 0, stream>>>(t, nZero);

    precompute_e<<<NTOT / 4, 256, 0, stream>>>(ego, Wu, du, pu, Wi, di, pi, e);

    dim3 ggrid(NTOT / 64, 1, KSPLIT);
    gemm64<true ><<<ggrid, 128, 0, stream>>>(adj, e, t);   // t = adj^T @ e
    gemm64<false><<<ggrid, 128, 0, stream>>>(adj, t, h);   // h = adj   @ t

    ln_residual<<<NTOT / 4, 256, 0, stream>>>(h, ego, lg, lb, out);
}